// NeuroSampler_9234179687018
// MI455X (gfx1250) — compile-verified
//
#include <hip/hip_runtime.h>
#include <hip/hip_bf16.h>
#include <math.h>

// ---------------------------------------------------------------------------
// Top-p (nucleus) sampling, sort-free, one workgroup per row.
//   B=256 rows, V=128000 vocab, TOP_P=0.9, temperature=1.
// Pass A: row max via 8-deep async global->LDS pipeline (HBM-bound, 131 MB).
// Pass B: exp-mass histogram (ds_add_f32) + parallel scan -> threshold tau.
// Pass C: Gumbel-max argmax over {x >= tau} == categorical over nucleus (L2).
// ---------------------------------------------------------------------------

#define THREADS   256
#define VOCAB     128000
#define CHUNK     (THREADS * 4)        // 1024 floats per block-chunk (16B/lane)
#define NCHUNKS   (VOCAB / CHUNK)      // 125, exact
#define DEPTH     8                    // async pipeline stages (32 KB staging)
#define NBINS     2048
#define SEG       (NBINS / THREADS)    // 8 bins per thread in the scan
#define BIN_RANGE 16.0f                // logit range below max covered by bins
#define TOP_P_C   0.9f

// ---- CDNA5 async global->LDS (inline asm: portable across both toolchains) --
__device__ __forceinline__ unsigned lds_off_u32(const void* p) {
  // generic pointer to LDS: low 32 bits are the LDS byte address (aperture map)
  return (unsigned)(unsigned long long)p;
}
__device__ __forceinline__ void async_ld_b128(unsigned lds_off, const float* g) {
  asm volatile("global_load_async_to_lds_b128 %0, %1, off"
               :: "v"(lds_off), "v"(g) : "memory");
}
__device__ __forceinline__ void wait_async_drain() {
  asm volatile("s_wait_asynccnt 0x0" ::: "memory");
}
__device__ __forceinline__ void wait_async_steady() {   // oldest of DEPTH done
  asm volatile("s_wait_asynccnt 0x7" ::: "memory");
}
__device__ __forceinline__ void wait_ds0() {
  asm volatile("s_wait_dscnt 0x0" ::: "memory");
}

// counter-based PCG-style hash -> deterministic per (row, token) RNG
__device__ __forceinline__ unsigned pcg_hash(unsigned x) {
  x = x * 747796405u + 2891336453u;
  x = ((x >> ((x >> 28u) + 4u)) ^ x) * 277803737u;
  return (x >> 22u) ^ x;
}

__global__ void __launch_bounds__(THREADS)
topp_sample_kernel(const float* __restrict__ logits,
                   const float* __restrict__ bias,
                   float* __restrict__ out)
{
  __shared__ __align__(16) float stage[DEPTH][CHUNK];  // 32 KB async staging
  __shared__ float hist[NBINS];                        // 8 KB
  __shared__ float red_f[THREADS];
  __shared__ int   red_i[THREADS];
  __shared__ float s_bcast[2];

  const int tid = threadIdx.x;
  const int row = blockIdx.x;
  const float* rowp = logits + (size_t)row * VOCAB;

  // ---------------- Pass A: row max of (logit + bias), 8-deep async ---------
  // Prologue: fill the pipeline.
#pragma unroll
  for (int p = 0; p < DEPTH; ++p)
    async_ld_b128(lds_off_u32(&stage[p][tid * 4]),
                  rowp + (size_t)p * CHUNK + tid * 4);

  // zero histogram while the pipeline is in flight (disjoint LDS region)
  for (int i = tid; i < NBINS; i += THREADS) hist[i] = 0.0f;

  float m = -INFINITY;
  // Steady state: ASYNCcnt == DEPTH outstanding; wait <=7 means oldest landed.
  for (int c = 0; c < NCHUNKS - DEPTH; ++c) {
    wait_async_steady();
    const float4 bb = *(const float4*)(bias + (size_t)c * CHUNK + tid * 4);
    const float4 xv = *(const float4*)(&stage[c & (DEPTH - 1)][tid * 4]);
    m = fmaxf(m, fmaxf(fmaxf(xv.x + bb.x, xv.y + bb.y),
                       fmaxf(xv.z + bb.z, xv.w + bb.w)));
    wait_ds0();  // LDS reads retired before async rewrite of this buffer
    async_ld_b128(lds_off_u32(&stage[c & (DEPTH - 1)][tid * 4]),
                  rowp + (size_t)(c + DEPTH) * CHUNK + tid * 4);
  }
  // Drain: everything outstanding has landed; consume last DEPTH buffers.
  wait_async_drain();
  for (int c = NCHUNKS - DEPTH; c < NCHUNKS; ++c) {
    const float4 bb = *(const float4*)(bias + (size_t)c * CHUNK + tid * 4);
    const float4 xv = *(const float4*)(&stage[c & (DEPTH - 1)][tid * 4]);
    m = fmaxf(m, fmaxf(fmaxf(xv.x + bb.x, xv.y + bb.y),
                       fmaxf(xv.z + bb.z, xv.w + bb.w)));
  }
  red_f[tid] = m;
  __syncthreads();
  for (int s = THREADS / 2; s > 0; s >>= 1) {
    if (tid < s) red_f[tid] = fmaxf(red_f[tid], red_f[tid + s]);
    __syncthreads();
  }
  m = red_f[0];
  __syncthreads();

  // ---------------- Pass B: exp-mass histogram (L2-resident reads) ----------
  const float inv_bw = (float)NBINS / BIN_RANGE;
#pragma unroll 2
  for (int c = 0; c < NCHUNKS; ++c) {
    const int v0 = c * CHUNK + tid * 4;
    const float4 xv = *(const float4*)(rowp + v0);
    const float4 bb = *(const float4*)(bias + v0);
    float xs[4] = {xv.x + bb.x, xv.y + bb.y, xv.z + bb.z, xv.w + bb.w};
#pragma unroll
    for (int k = 0; k < 4; ++k) {
      const float e = __expf(xs[k] - m);
      int b = (int)((m - xs[k]) * inv_bw);
      b = b < (NBINS - 1) ? b : (NBINS - 1);   // far tail -> last bin (e ~ 1e-7)
      atomicAdd(&hist[b], e);                  // ds_add_f32
    }
  }
  __syncthreads();

  // Parallel segmented scan over bins: Z and the crossing bin in O(log N).
  float seg_sum = 0.0f;
#pragma unroll
  for (int k = 0; k < SEG; ++k) seg_sum += hist[tid * SEG + k];
  red_f[tid] = seg_sum;
  __syncthreads();
  for (int off = 1; off < THREADS; off <<= 1) {   // Hillis-Steele inclusive scan
    const float v = (tid >= off) ? red_f[tid - off] : 0.0f;
    __syncthreads();
    red_f[tid] += v;
    __syncthreads();
  }
  const float Z      = red_f[THREADS - 1];
  const float target = TOP_P_C * Z;
  const float incl   = red_f[tid];
  const float excl   = incl - seg_sum;
  if (tid == 0) s_bcast[0] = m - BIN_RANGE;       // fallback threshold
  __syncthreads();
  if (excl < target && target <= incl) {          // unique bracketing segment
    float cum = excl;
    int b = tid * SEG + SEG - 1;
#pragma unroll
    for (int k = 0; k < SEG; ++k) {
      cum += hist[tid * SEG + k];
      if (cum >= target) { b = tid * SEG + k; break; }
    }
    s_bcast[0] = m - (float)(b + 1) * (BIN_RANGE / (float)NBINS);
  }
  __syncthreads();
  const float tau = s_bcast[0];
  __syncthreads();

  // ---------------- Pass C: Gumbel-max over the nucleus (L2-resident) -------
  const unsigned seed = 0x9E3779B9u * (unsigned)(row + 1) ^ 42u;
  float best = -INFINITY;
  int besti = 0;
#pragma unroll 2
  for (int c = 0; c < NCHUNKS; ++c) {
    const int v0 = c * CHUNK + tid * 4;
    const float4 xv = *(const float4*)(rowp + v0);
    const float4 bb = *(const float4*)(bias + v0);
    float xs[4] = {xv.x + bb.x, xv.y + bb.y, xv.z + bb.z, xv.w + bb.w};
#pragma unroll
    for (int k = 0; k < 4; ++k) {
      if (xs[k] >= tau) {
        const unsigned h = pcg_hash(seed ^ ((unsigned)(v0 + k) * 0x85EBCA6Bu));
        const float u = ((float)(h >> 8) + 0.5f) * (1.0f / 16777216.0f); // (0,1)
        const float g = -__logf(-__logf(u));
        const float sc = xs[k] + g;
        if (sc > best) { best = sc; besti = v0 + k; }
      }
    }
  }
  red_f[tid] = best;
  red_i[tid] = besti;
  __syncthreads();
  for (int s = THREADS / 2; s > 0; s >>= 1) {
    if (tid < s && red_f[tid + s] > red_f[tid]) {
      red_f[tid] = red_f[tid + s];
      red_i[tid] = red_i[tid + s];
    }
    __syncthreads();
  }
  if (tid == 0) out[row] = (float)red_i[0];
}

extern "C" void kernel_launch(void* const* d_in, const int* in_sizes, int n_in,
                              void* d_out, int out_size, void* d_ws, size_t ws_size,
                              hipStream_t stream) {
  (void)n_in; (void)d_ws; (void)ws_size; (void)out_size;
  const float* logits = (const float*)d_in[0];   // (B, V) fp32
  const float* bias   = (const float*)d_in[1];   // (V,)  fp32
  float* out = (float*)d_out;                    // (B,) token ids as float
  const int V = in_sizes[1];
  const int B = in_sizes[0] / V;                 // 256
  topp_sample_kernel<<<B, THREADS, 0, stream>>>(logits, bias, out);
}